// GraphTokenRelationshipModel_8031588843863
// MI455X (gfx1250) — compile-verified
//
#include <hip/hip_runtime.h>
#include <hip/hip_bf16.h>

#define NEG_SLOPE 0.2f

typedef float v2f __attribute__((ext_vector_type(2)));
typedef float v8f __attribute__((ext_vector_type(8)));

// ---------------------------------------------------------------------------
// WMMA f32 GEMM: C[M,Nc] = A[M,K] * B[K,Nc], optional fused bias+relu.
// Compile-time K (fully unrolled K-loop). One block = 8 waves = one 16-col
// N tile x 8 M tiles; B tile staged transposed in LDS and shared by all waves.
// Per-WMMA data movement: 1x global_load_b64 (A) + 1x ds_load_b64 (B).
//
// Fragment layouts (V_WMMA_F32_16X16X4_F32, wave32):
//   A 16x4 : lane m = lane&15, hi = lane>>4 -> a = A[m][2*hi + {0,1}]
//   B 4x16 : lane n = lane&15               -> b = B[2*hi + {0,1}][n]
//   C 16x16: vgpr r, lanes 0-15 -> C[r][n], lanes 16-31 -> C[r+8][n]
// ---------------------------------------------------------------------------
template <int K>
__global__ __launch_bounds__(256) void gemm_wmma_f32(
    const float* __restrict__ A, const float* __restrict__ B,
    float* __restrict__ C, int M, int Nc,
    const float* __restrict__ bias, int fuse_relu_bias) {
  __shared__ float Blds[16 * K];

  const int ntN = Nc >> 4;
  const int tn = blockIdx.x % ntN;
  const int tmb = (blockIdx.x / ntN) * 8;
  const int wid = threadIdx.x >> 5;
  const int lane = threadIdx.x & 31;
  const int col0 = tn * 16;

  // cooperative stage of B tile, transposed: Blds[c*K + k] = B[k][col0+c]
  for (int idx = threadIdx.x; idx < 16 * K; idx += 256) {
    const int k = idx >> 4;
    const int c = idx & 15;
    Blds[c * K + k] = B[(size_t)k * Nc + col0 + c];
  }
  __syncthreads();

  const int tm = tmb + wid;
  if (tm * 16 >= M) return;  // uniform per wave; EXEC all-ones for WMMA

  const int mn = lane & 15;  // A row / B-C col within tile
  const int hi = lane >> 4;  // K-pair selector
  const float* Arow = A + (size_t)(tm * 16 + mn) * K;
  const float* Bcol = Blds + mn * K;

  v8f acc = {};
#pragma unroll
  for (int k = 0; k < K; k += 4) {
    const int ka = k + 2 * hi;
    const v2f a = *(const v2f*)(Arow + ka);   // global_load_b64
    const v2f b = *(const v2f*)(Bcol + ka);   // ds_load_b64
    // 8 args: (neg_a, A, neg_b, B, c_mod, C, reuse_a, reuse_b)
    acc = __builtin_amdgcn_wmma_f32_16x16x4_f32(
        false, a, false, b, (short)0, acc, false, false);
  }

  const int ncol = col0 + mn;
  const float bv = fuse_relu_bias ? bias[ncol] : 0.0f;
#pragma unroll
  for (int r = 0; r < 8; ++r) {
    const int row = tm * 16 + r + 8 * hi;
    float v = acc[r];
    if (fuse_relu_bias) {
      v += bv;
      v = v > 0.0f ? v : 0.0f;
    }
    C[(size_t)row * Nc + ncol] = v;
  }
}

// ---------------------------------------------------------------------------
// Zero-pad embed operands so the GEMM runs the clean K=64 path.
// ---------------------------------------------------------------------------
__global__ __launch_bounds__(256) void pad_embed_A(
    const float* __restrict__ nf, float* __restrict__ Apad, int Nn) {
  const int t = blockIdx.x * blockDim.x + threadIdx.x;
  if (t >= Nn * 64) return;
  const int n = t >> 6, k = t & 63;
  Apad[t] = (k < 50) ? nf[(size_t)n * 50 + k] : 0.0f;
}

__global__ __launch_bounds__(256) void pad_embed_B(
    const float* __restrict__ W, float* __restrict__ Bpad) {
  const int t = blockIdx.x * blockDim.x + threadIdx.x;
  if (t >= 64 * 128) return;
  const int k = t >> 7, c = t & 127;
  Bpad[t] = (k < 50) ? W[(size_t)k * 128 + c] : 0.0f;
}

// ---------------------------------------------------------------------------
// Per-(node,head) attention logits: al_s[n,h] = <h[n,h,:], att_src[h,:]>
// ---------------------------------------------------------------------------
__global__ __launch_bounds__(256) void attn_logits_kernel(
    const float* __restrict__ h, const float* __restrict__ att_s,
    const float* __restrict__ att_d, float* __restrict__ al_s,
    float* __restrict__ al_d, int Nn) {
  const int t = blockIdx.x * blockDim.x + threadIdx.x;
  if (t >= Nn * 4) return;
  const int n = t >> 2, hd = t & 3;
  const float* hp = h + (size_t)n * 128 + hd * 32;
  const float* as = att_s + hd * 32;
  const float* ad = att_d + hd * 32;
  float s = 0.0f, d = 0.0f;
#pragma unroll
  for (int i = 0; i < 32; ++i) {
    const float v = hp[i];
    s = fmaf(v, as[i], s);
    d = fmaf(v, ad[i], d);
  }
  al_s[t] = s;
  al_d[t] = d;
}

__device__ __forceinline__ float leaky(float v) {
  return v > 0.0f ? v : v * NEG_SLOPE;
}
// monotonic float -> int key for integer atomicMax
__device__ __forceinline__ int fkey(float f) {
  int k = __float_as_int(f);
  return k >= 0 ? k : (k ^ 0x7fffffff);
}
__device__ __forceinline__ float fkey_inv(int k) {
  return __int_as_float(k >= 0 ? k : (k ^ 0x7fffffff));
}

// ---------------------------------------------------------------------------
// Per-layer init: acc = 0, denom = 0, emax = below key(-inf)
// ---------------------------------------------------------------------------
__global__ __launch_bounds__(256) void layer_init_kernel(
    float* __restrict__ acc, float* __restrict__ denom, int* __restrict__ emax,
    int Nn) {
  const int t = blockIdx.x * blockDim.x + threadIdx.x;
  if (t < Nn * 128) acc[t] = 0.0f;
  if (t < Nn * 4) {
    denom[t] = 0.0f;
    emax[t] = (int)0x80000000;
  }
}

// ---------------------------------------------------------------------------
// Pass 1 over edges: segment max of leaky(al_s[src]+al_d[dst]) per (dst,head)
// ---------------------------------------------------------------------------
__global__ __launch_bounds__(256) void edge_max_kernel(
    const int* __restrict__ src, const int* __restrict__ dst,
    const float* __restrict__ al_s, const float* __restrict__ al_d,
    int* __restrict__ emax, int E, int Nn) {
  const int e = blockIdx.x * blockDim.x + threadIdx.x;
  if (e >= E + Nn) return;
  int s_, d_;
  if (e < E) { s_ = src[e]; d_ = dst[e]; } else { s_ = d_ = e - E; }
  const int s4 = s_ * 4, d4 = d_ * 4;
#pragma unroll
  for (int hd = 0; hd < 4; ++hd) {
    const float v = leaky(al_s[s4 + hd] + al_d[d4 + hd]);
    atomicMax(&emax[d4 + hd], fkey(v));
  }
}

// ---------------------------------------------------------------------------
// Pass 2 over edges: ex = exp(e - emax[dst]); store per edge; denom += ex
// ---------------------------------------------------------------------------
__global__ __launch_bounds__(256) void edge_expsum_kernel(
    const int* __restrict__ src, const int* __restrict__ dst,
    const float* __restrict__ al_s, const float* __restrict__ al_d,
    const int* __restrict__ emax, float* __restrict__ denom,
    float* __restrict__ exb, int E, int Nn) {
  const int e = blockIdx.x * blockDim.x + threadIdx.x;
  if (e >= E + Nn) return;
  int s_, d_;
  if (e < E) { s_ = src[e]; d_ = dst[e]; } else { s_ = d_ = e - E; }
  const int s4 = s_ * 4, d4 = d_ * 4;
#pragma unroll
  for (int hd = 0; hd < 4; ++hd) {
    const float v = leaky(al_s[s4 + hd] + al_d[d4 + hd]);
    const float mx = fkey_inv(emax[d4 + hd]);
    const float ex = __expf(v - mx);
    exb[(size_t)e * 4 + hd] = ex;
    unsafeAtomicAdd(&denom[d4 + hd], ex);
  }
}

// ---------------------------------------------------------------------------
// Pass 3 over edges: acc[dst, h, :] += ex * h[src, h, :]
// one wave per edge, each lane carries 4 consecutive floats (float4 gather)
// ---------------------------------------------------------------------------
__global__ __launch_bounds__(256) void edge_aggregate_kernel(
    const int* __restrict__ src, const int* __restrict__ dst,
    const float* __restrict__ h, const float* __restrict__ exb,
    float* __restrict__ acc, int E, int Nn) {
  const int t = blockIdx.x * blockDim.x + threadIdx.x;
  const int e = t >> 5;
  const int lane = t & 31;
  if (e >= E + Nn) return;
  int s_, d_;
  if (e < E) { s_ = src[e]; d_ = dst[e]; } else { s_ = d_ = e - E; }
  const int d0 = lane * 4;   // offset within 128 = head*32 + dim
  const int hd = lane >> 3;  // 4 consecutive dims never cross a head
  const float w = exb[(size_t)e * 4 + hd];
  const float4 hv = *(const float4*)(h + (size_t)s_ * 128 + d0);
  float* ap = acc + (size_t)d_ * 128 + d0;
  unsafeAtomicAdd(ap + 0, hv.x * w);
  unsafeAtomicAdd(ap + 1, hv.y * w);
  unsafeAtomicAdd(ap + 2, hv.z * w);
  unsafeAtomicAdd(ap + 3, hv.w * w);
}

// ---------------------------------------------------------------------------
// Per-node finalize: x = relu(acc / denom[head] + bias)  (head-concat layout)
// ---------------------------------------------------------------------------
__global__ __launch_bounds__(256) void layer_finalize_kernel(
    const float* __restrict__ acc, const float* __restrict__ denom,
    const float* __restrict__ bias, float* __restrict__ x, int Nn) {
  const int t = blockIdx.x * blockDim.x + threadIdx.x;
  if (t >= Nn * 128) return;
  const int n = t >> 7;
  const int c = t & 127;
  const int hd = c >> 5;
  const float dn = denom[n * 4 + hd];
  float v = acc[t] / dn + bias[c];
  x[t] = v > 0.0f ? v : 0.0f;
}

// ---------------------------------------------------------------------------
// Head: out[n] = sigmoid(<hidden[n,:64], W2> + b2)
// ---------------------------------------------------------------------------
__global__ __launch_bounds__(256) void head_kernel(
    const float* __restrict__ hidden, const float* __restrict__ W2,
    const float* __restrict__ b2, float* __restrict__ out, int Nn) {
  const int n = blockIdx.x * blockDim.x + threadIdx.x;
  if (n >= Nn) return;
  const float* hp = hidden + (size_t)n * 64;
  float a = b2[0];
#pragma unroll
  for (int i = 0; i < 64; ++i) a = fmaf(hp[i], W2[i], a);
  out[n] = 1.0f / (1.0f + __expf(-a));
}

// ---------------------------------------------------------------------------

extern "C" void kernel_launch(void* const* d_in, const int* in_sizes, int n_in,
                              void* d_out, int out_size, void* d_ws,
                              size_t ws_size, hipStream_t stream) {
  const float* nf      = (const float*)d_in[0];   // [N, 50]
  const int*   ei      = (const int*)d_in[1];     // [2, E]
  const float* W_embed = (const float*)d_in[2];   // [50, 128]
  const float* b_embed = (const float*)d_in[3];   // [128]
  const float* Wg      = (const float*)d_in[4];   // [3, 128, 128]
  const float* att_src = (const float*)d_in[5];   // [3, 4, 32]
  const float* att_dst = (const float*)d_in[6];   // [3, 4, 32]
  const float* bias_g  = (const float*)d_in[7];   // [3, 128]
  const float* W1      = (const float*)d_in[8];   // [128, 64]
  const float* b1      = (const float*)d_in[9];   // [64]
  const float* W2      = (const float*)d_in[10];  // [64, 1]
  const float* b2      = (const float*)d_in[11];  // [1]
  float* out = (float*)d_out;

  const int Nn = in_sizes[0] / 50;  // 50000
  const int Hid = 128;
  const int Eg = in_sizes[1] / 2;   // 800000
  const int Etot = Eg + Nn;         // + self loops

  // workspace layout (floats)
  float* ws    = (float*)d_ws;
  float* xbuf  = ws;                           // [N,128]
  float* hbuf  = xbuf + (size_t)Nn * Hid;      // [N,128] (reused: Bpad, hidden)
  float* accb  = hbuf + (size_t)Nn * Hid;      // [N,128] (reused: Apad)
  float* al_s  = accb + (size_t)Nn * Hid;      // [N,4]
  float* al_d  = al_s + (size_t)Nn * 4;        // [N,4]
  float* denom = al_d + (size_t)Nn * 4;        // [N,4]
  int*   emax  = (int*)(denom + (size_t)Nn * 4);   // [N,4]
  float* exb   = (float*)(emax + (size_t)Nn * 4);  // [Etot,4]

  const int* srcI = ei;
  const int* dstI = ei + Eg;

  const int mtiles = (Nn + 15) / 16;        // 3125
  const int mblocks = (mtiles + 7) / 8;     // 391

  // 1) embed: pad to K=64, then x = relu(Apad @ Bpad + b_embed)
  {
    float* Apad = accb;  // dead until layer_init
    float* Bpad = hbuf;  // dead until layer GEMM output
    pad_embed_A<<<(Nn * 64 + 255) / 256, 256, 0, stream>>>(nf, Apad, Nn);
    pad_embed_B<<<(64 * 128 + 255) / 256, 256, 0, stream>>>(W_embed, Bpad);
    gemm_wmma_f32<64><<<mblocks * (Hid / 16), 256, 0, stream>>>(
        Apad, Bpad, xbuf, Nn, Hid, b_embed, 1);
  }

  // 2) GAT layers
  for (int l = 0; l < 3; ++l) {
    layer_init_kernel<<<(Nn * 128 + 255) / 256, 256, 0, stream>>>(
        accb, denom, emax, Nn);

    gemm_wmma_f32<128><<<mblocks * (Hid / 16), 256, 0, stream>>>(
        xbuf, Wg + (size_t)l * Hid * Hid, hbuf, Nn, Hid, nullptr, 0);

    attn_logits_kernel<<<(Nn * 4 + 255) / 256, 256, 0, stream>>>(
        hbuf, att_src + l * 128, att_dst + l * 128, al_s, al_d, Nn);

    edge_max_kernel<<<(Etot + 255) / 256, 256, 0, stream>>>(
        srcI, dstI, al_s, al_d, emax, Eg, Nn);

    edge_expsum_kernel<<<(Etot + 255) / 256, 256, 0, stream>>>(
        srcI, dstI, al_s, al_d, emax, denom, exb, Eg, Nn);

    edge_aggregate_kernel<<<((size_t)Etot * 32 + 255) / 256, 256, 0, stream>>>(
        srcI, dstI, hbuf, exb, accb, Eg, Nn);

    layer_finalize_kernel<<<(Nn * 128 + 255) / 256, 256, 0, stream>>>(
        accb, denom, bias_g + l * Hid, xbuf, Nn);
  }

  // 3) MLP: hidden = relu(x @ W1 + b1)   (Nc=64)
  gemm_wmma_f32<128><<<mblocks * (64 / 16), 256, 0, stream>>>(
      xbuf, W1, hbuf, Nn, 64, b1, 1);

  // 4) out = sigmoid(hidden @ W2 + b2)
  head_kernel<<<(Nn + 255) / 256, 256, 0, stream>>>(hbuf, W2, b2, out, Nn);
}